// LeadingZeroDetector48_23407571764134
// MI455X (gfx1250) — compile-verified
//
#include <hip/hip_runtime.h>
#include <stdint.h>

#define AS1 __attribute__((address_space(1)))
#define AS3 __attribute__((address_space(3)))

typedef __attribute__((ext_vector_type(2))) float    v2f;
typedef __attribute__((ext_vector_type(4))) uint32_t v4u;
typedef __attribute__((ext_vector_type(4))) int      v4i;

#ifndef __has_builtin
#define __has_builtin(x) 0
#endif

#if __has_builtin(__builtin_amdgcn_global_load_async_to_lds_b128)
#define LZC_USE_ASYNC 1
#else
#define LZC_USE_ASYNC 0
#endif

static __device__ __forceinline__ void lzc_wait_async0() {
#if __has_builtin(__builtin_amdgcn_s_wait_asynccnt)
  __builtin_amdgcn_s_wait_asynccnt(0);
#else
  asm volatile("s_wait_asynccnt 0" ::: "memory");
#endif
}

static __device__ __forceinline__ void async_cp16(const v4u* g, v4u* l) {
#if LZC_USE_ASYNC
  AS1 v4i* gq = (AS1 v4i*)(uintptr_t)g;
  AS3 v4i* lp = (AS3 v4i*)(uint32_t)(uintptr_t)l;
  __builtin_amdgcn_global_load_async_to_lds_b128(gq, lp, 0, 0);
#else
  *l = *g;
#endif
}

// One block = 256 threads = 256 words of 48 floats (48 KB staged via LDS).
// Input floats are exactly 0.0f (0x00000000) or 1.0f (0x3F800000); bit 23
// of the raw pattern IS the logical bit, so the soft-logic LZC reduces to an
// exact integer count-trailing-zeros over a 48-bit packed mask.
__global__ __launch_bounds__(256) void lzc48_kernel(const uint32_t* __restrict__ X,
                                                    float* __restrict__ out,
                                                    int nwords) {
  __shared__ v4u lds[12 * 256];  // 48 KB

  const int t = threadIdx.x;
  const long long blockWordBase = (long long)blockIdx.x * 256;
  const long long totalVec = (long long)nwords * 12;   // number of 16B chunks
  const long long vecBase  = blockWordBase * 12;

  const v4u* gbase = (const v4u*)X + vecBase;

  // ---- Stage 48 KB, fully coalesced (lane L fetches chunk k*256+L) ----
  if (vecBase + 12 * 256 <= totalVec) {
    // Fast path (uniform branch; all blocks for the reference shape):
    // one 64-bit base, constant byte offsets k*4096 foldable into the insn.
    const v4u* gp = gbase + t;
#pragma unroll
    for (int k = 0; k < 12; ++k) {
      async_cp16(gp + k * 256, &lds[k * 256 + t]);
    }
  } else {
    // Tail block: clamp each chunk index (duplicate loads are discarded).
#pragma unroll
    for (int k = 0; k < 12; ++k) {
      long long gi = vecBase + (long long)k * 256 + t;
      if (gi >= totalVec) gi = totalVec - 1;
      async_cp16((const v4u*)X + gi, &lds[k * 256 + t]);
    }
  }
#if LZC_USE_ASYNC
  lzc_wait_async0();
#endif
  __syncthreads();

  // ---- Each thread scans its own word out of LDS (12 x ds_load_b128) ----
  uint32_t lo = 0u, hi = 0u;
#pragma unroll
  for (int k = 0; k < 8; ++k) {
    v4u v = lds[t * 12 + k];
    uint32_t nib = ((v.x >> 23) & 1u) | ((v.y >> 22) & 2u) |
                   ((v.z >> 21) & 4u) | ((v.w >> 20) & 8u);
    lo |= nib << (4 * k);
  }
#pragma unroll
  for (int k = 8; k < 12; ++k) {
    v4u v = lds[t * 12 + k];
    uint32_t nib = ((v.x >> 23) & 1u) | ((v.y >> 22) & 2u) |
                   ((v.z >> 21) & 4u) | ((v.w >> 20) & 8u);
    hi |= nib << (4 * (k - 8));
  }

  uint32_t lzc;
  if (lo)      lzc = (uint32_t)__builtin_ctz(lo);
  else if (hi) lzc = 32u + (uint32_t)__builtin_ctz(hi);
  else         lzc = 48u;                 // all-zero word -> 0b110000

  // ---- Emit 6 MSB-first bits as three aligned float2 NT stores ----
  const long long w = blockWordBase + t;
  if (w < (long long)nwords) {
    v2f a, b, c;
    a.x = (float)((lzc >> 5) & 1u);
    a.y = (float)((lzc >> 4) & 1u);
    b.x = (float)((lzc >> 3) & 1u);
    b.y = (float)((lzc >> 2) & 1u);
    c.x = (float)((lzc >> 1) & 1u);
    c.y = (float)( lzc        & 1u);
    v2f* o = (v2f*)(out + w * 6);         // w*6 floats -> 8B aligned
    __builtin_nontemporal_store(a, o + 0);
    __builtin_nontemporal_store(b, o + 1);
    __builtin_nontemporal_store(c, o + 2);
  }
}

extern "C" void kernel_launch(void* const* d_in, const int* in_sizes, int n_in,
                              void* d_out, int out_size, void* d_ws, size_t ws_size,
                              hipStream_t stream) {
  (void)n_in; (void)out_size; (void)d_ws; (void)ws_size;
  const uint32_t* X = (const uint32_t*)d_in[0];
  float* out = (float*)d_out;
  const int nwords = in_sizes[0] / 48;              // 2,097,152 for the ref shapes
  const int blocks = (nwords + 255) / 256;
  lzc48_kernel<<<dim3(blocks), dim3(256), 0, stream>>>(X, out, nwords);
}